// GNN_9689446219777
// MI455X (gfx1250) — compile-verified
//
#include <hip/hip_runtime.h>
#include <math.h>

// ---------------- problem constants ----------------
#define N_NODES 116
#define N_EDGES 1160
#define NCH     256                 // channels after grouped conv
#define F       653                 // SAGE feature dim
#define FP      672                 // padded feature dim (21*32: N splits into 64x32 wave tiles)
#define ROWS    (N_NODES * NCH)     // 29696 "rows" = (node, channel) pairs
#define KTOT    (2 * FP)            // fused K for [agg | x] GEMM

typedef __attribute__((ext_vector_type(2))) float v2f;
typedef __attribute__((ext_vector_type(8))) float v8f;

// ---------------- grouped Conv1d (64->256, k=5, groups=4) + bias, pad to FP ----
__global__ __launch_bounds__(256) void conv1d_grouped(
    const float* __restrict__ x,     // [116,64,657]
    const float* __restrict__ w,     // [256,16,5]
    const float* __restrict__ b,     // [256]
    float* __restrict__ H)           // [ROWS, FP]
{
    __shared__ float wsh[80];
    __shared__ float bsh;
    const int row = blockIdx.x;            // n*256 + o
    const int n = row >> 8, o = row & 255;
    if (threadIdx.x < 80) wsh[threadIdx.x] = w[(size_t)o * 80 + threadIdx.x];
    if (threadIdx.x == 0) bsh = b[o];
    __syncthreads();
    const float* xn = x + (size_t)n * 64 * 657 + (size_t)(o >> 6) * 16 * 657;
    for (int p = threadIdx.x; p < FP; p += blockDim.x) {
        if (p < F) {
            float acc = bsh;
#pragma unroll
            for (int i = 0; i < 16; i++) {
                const float* xr = xn + i * 657 + p;
#pragma unroll
                for (int k = 0; k < 5; k++) acc = fmaf(wsh[i * 5 + k], xr[k], acc);
            }
            H[(size_t)row * FP + p] = acc;
        } else {
            H[(size_t)row * FP + p] = 0.0f;   // keep pad columns exactly zero
        }
    }
}

// ---------------- build per-destination in-edge lists (CSR-ish) ----------------
__global__ void build_csr(const long long* __restrict__ ei,  // [2, N_EDGES] int64
                          int* __restrict__ deg, int* __restrict__ lists)
{
    const int n = blockIdx.x * blockDim.x + threadIdx.x;
    if (n >= N_NODES) return;
    int d = 0;
    for (int e = 0; e < N_EDGES; e++) {
        if ((int)ei[N_EDGES + e] == n)          // dst == n
            lists[(size_t)n * N_EDGES + d++] = (int)ei[e];  // src
    }
    deg[n] = d;
}

// -------- transpose+stack SAGE weights: Wt[f][g] (K x N, zero padded) ----------
__global__ void build_wt(const float* __restrict__ wl, const float* __restrict__ wr,
                         const float* __restrict__ sb,
                         float* __restrict__ Wt, float* __restrict__ biasP)
{
    const int f = blockIdx.x;                  // 0..KTOT-1
    const int s = (f >= FP) ? 1 : 0;
    const int fl = s ? (f - FP) : f;
    const float* W = s ? wr : wl;
    for (int g = threadIdx.x; g < FP; g += blockDim.x) {
        float v = 0.0f;
        if (fl < F && g < F) v = W[(size_t)g * F + fl];   // Wt[f][g] = W[g][f]
        Wt[(size_t)f * FP + g] = v;
    }
    if (f == 0)
        for (int g = threadIdx.x; g < FP; g += blockDim.x)
            biasP[g] = (g < F) ? sb[g] : 0.0f;
}

// ---------------- segment-max aggregation (dst-parallel, L2-resident) ----------
__global__ __launch_bounds__(256) void sage_aggmax(
    const float* __restrict__ H, const int* __restrict__ deg,
    const int* __restrict__ lists, float* __restrict__ Agg)
{
    const int row = blockIdx.x;                // n*256 + c
    const int node = row >> 8;
    const int c = row & 255;
    const int d = deg[node];
    const int* lst = lists + (size_t)node * N_EDGES;
    for (int f = threadIdx.x; f < FP; f += blockDim.x) {
        float m = -INFINITY;
        for (int e = 0; e < d; e++) {
            const int s = lst[e];
            m = fmaxf(m, H[((size_t)(s << 8) + c) * FP + f]);
        }
        Agg[(size_t)row * FP + f] = (d == 0) ? 0.0f : m;   // empty segment -> 0
    }
}

// ------- fused SAGE GEMM: Out = Agg*Wl^T + X*Wr^T + b  via V_WMMA_F32_16X16X4 --
// grid = (ROWS/512, FP/32); block = 256 (8 waves); wave owns a 64x32 tile
// (4 M-subtiles x 2 N-subtiles -> 8 WMMAs per k-step, 8 VMEM ops per k-step).
__global__ __launch_bounds__(256) void sage_gemm(
    const float* __restrict__ Agg, const float* __restrict__ Xin,
    const float* __restrict__ Wt,  const float* __restrict__ biasP,
    float* __restrict__ Out)
{
    const int lane = threadIdx.x & 31;
    const int wave = threadIdx.x >> 5;
    const int half = lane >> 4;        // 0: K lanes {0,1}, 1: K lanes {2,3}
    const int lm   = lane & 15;
    const int mBase = blockIdx.x * 512 + wave * 64;
    const int nBase = blockIdx.y * 32;

    v8f acc00 = {}, acc01 = {}, acc10 = {}, acc11 = {};
    v8f acc20 = {}, acc21 = {}, acc30 = {}, acc31 = {};

    for (int s = 0; s < 2; s++) {
        const float* A = s ? Xin : Agg;
        const float* B = Wt + (size_t)s * FP * FP;
#pragma unroll 2
        for (int k = 0; k < FP; k += 4) {
            const int ka = k + half * 2;
            // B fragments: rows ka, ka+1 at columns nBase+lm and nBase+16+lm
            v2f bf0, bf1;
            bf0.x = B[(size_t)ka * FP + nBase + lm];
            bf0.y = B[(size_t)(ka + 1) * FP + nBase + lm];
            bf1.x = B[(size_t)ka * FP + nBase + 16 + lm];
            bf1.y = B[(size_t)(ka + 1) * FP + nBase + 16 + lm];
            // A fragments for 4 M-subtiles: row (mBase + t*16 + lm), cols ka..ka+1
            const float2 a0 = *(const float2*)(A + (size_t)(mBase +  0 + lm) * FP + ka);
            const float2 a1 = *(const float2*)(A + (size_t)(mBase + 16 + lm) * FP + ka);
            const float2 a2 = *(const float2*)(A + (size_t)(mBase + 32 + lm) * FP + ka);
            const float2 a3 = *(const float2*)(A + (size_t)(mBase + 48 + lm) * FP + ka);
            v2f af0 = {a0.x, a0.y}, af1 = {a1.x, a1.y};
            v2f af2 = {a2.x, a2.y}, af3 = {a3.x, a3.y};
            acc00 = __builtin_amdgcn_wmma_f32_16x16x4_f32(false, af0, false, bf0, (short)0, acc00, false, false);
            acc01 = __builtin_amdgcn_wmma_f32_16x16x4_f32(false, af0, false, bf1, (short)0, acc01, false, false);
            acc10 = __builtin_amdgcn_wmma_f32_16x16x4_f32(false, af1, false, bf0, (short)0, acc10, false, false);
            acc11 = __builtin_amdgcn_wmma_f32_16x16x4_f32(false, af1, false, bf1, (short)0, acc11, false, false);
            acc20 = __builtin_amdgcn_wmma_f32_16x16x4_f32(false, af2, false, bf0, (short)0, acc20, false, false);
            acc21 = __builtin_amdgcn_wmma_f32_16x16x4_f32(false, af2, false, bf1, (short)0, acc21, false, false);
            acc30 = __builtin_amdgcn_wmma_f32_16x16x4_f32(false, af3, false, bf0, (short)0, acc30, false, false);
            acc31 = __builtin_amdgcn_wmma_f32_16x16x4_f32(false, af3, false, bf1, (short)0, acc31, false, false);
        }
    }

    const float bn0 = biasP[nBase + lm];
    const float bn1 = biasP[nBase + 16 + lm];
    // C/D layout: VGPR v -> (M = v + half*8, N = lm)
#pragma unroll
    for (int v = 0; v < 8; v++) {
        const int r = v + half * 8;
        Out[(size_t)(mBase +  0 + r) * FP + nBase + lm]      = acc00[v] + bn0;
        Out[(size_t)(mBase +  0 + r) * FP + nBase + 16 + lm] = acc01[v] + bn1;
        Out[(size_t)(mBase + 16 + r) * FP + nBase + lm]      = acc10[v] + bn0;
        Out[(size_t)(mBase + 16 + r) * FP + nBase + 16 + lm] = acc11[v] + bn1;
        Out[(size_t)(mBase + 32 + r) * FP + nBase + lm]      = acc20[v] + bn0;
        Out[(size_t)(mBase + 32 + r) * FP + nBase + 16 + lm] = acc21[v] + bn1;
        Out[(size_t)(mBase + 48 + r) * FP + nBase + lm]      = acc30[v] + bn0;
        Out[(size_t)(mBase + 48 + r) * FP + nBase + 16 + lm] = acc31[v] + bn1;
    }
}

// ---------------- mean pool over nodes ----------------
__global__ void pool_mean(const float* __restrict__ H, float* __restrict__ P)
{
    const int c = blockIdx.x;
    for (int f = threadIdx.x; f < FP; f += blockDim.x) {
        float s = 0.0f;
        for (int n = 0; n < N_NODES; n++) s += H[((size_t)(n * NCH + c)) * FP + f];
        P[(size_t)c * FP + f] = s * (1.0f / N_NODES);
    }
}

// ---------------- FC head ----------------
__device__ __forceinline__ float gelu_exact(float v) {
    return 0.5f * v * (1.0f + erff(v * 0.70710678118654752f));
}

__global__ void fc1_k(const float* __restrict__ P, const float* __restrict__ w,
                      const float* __restrict__ b, float* __restrict__ o)
{
    const int c = blockIdx.x, j = threadIdx.x;      // 128 threads
    const float* pr = P + (size_t)c * FP;
    const float* wr = w + (size_t)j * F;
    float s = b[j];
    for (int k = 0; k < F; k++) s = fmaf(pr[k], wr[k], s);
    o[c * 128 + j] = gelu_exact(s);
}

__global__ void fc2_k(const float* __restrict__ I, const float* __restrict__ w,
                      const float* __restrict__ b, float* __restrict__ o)
{
    const int c = blockIdx.x, j = threadIdx.x;      // 32 threads
    const float* ir = I + (size_t)c * 128;
    const float* wr = w + (size_t)j * 128;
    float s = b[j];
    for (int k = 0; k < 128; k++) s = fmaf(ir[k], wr[k], s);
    o[c * 32 + j] = gelu_exact(s);
}

__global__ void fc3_softmax(const float* __restrict__ I, const float* __restrict__ w,
                            const float* __restrict__ b, float* __restrict__ out)
{
    const int c = blockIdx.x, j = threadIdx.x;      // 4 threads
    __shared__ float lg[4];
    __shared__ float ex[4];
    const float* ir = I + (size_t)c * 32;
    const float* wr = w + (size_t)j * 32;
    float s = b[j];
    for (int k = 0; k < 32; k++) s = fmaf(ir[k], wr[k], s);
    lg[j] = s;
    __syncthreads();
    const float m = fmaxf(fmaxf(lg[0], lg[1]), fmaxf(lg[2], lg[3]));
    ex[j] = expf(s - m);
    __syncthreads();
    const float sum = ex[0] + ex[1] + ex[2] + ex[3];
    out[c * 4 + j] = ex[j] / sum;
}

// ---------------- launcher ----------------
extern "C" void kernel_launch(void* const* d_in, const int* in_sizes, int n_in,
                              void* d_out, int out_size, void* d_ws, size_t ws_size,
                              hipStream_t stream)
{
    const float*     x      = (const float*)d_in[0];
    const long long* edges  = (const long long*)d_in[1];   // int64 per reference
    const float*     conv_w = (const float*)d_in[2];
    const float*     conv_b = (const float*)d_in[3];
    const float*     wl     = (const float*)d_in[4];
    const float*     wr     = (const float*)d_in[5];
    const float*     sb     = (const float*)d_in[6];
    const float*     fc1w   = (const float*)d_in[7];
    const float*     fc1b   = (const float*)d_in[8];
    const float*     fc2w   = (const float*)d_in[9];
    const float*     fc2b   = (const float*)d_in[10];
    const float*     fc3w   = (const float*)d_in[11];
    const float*     fc3b   = (const float*)d_in[12];
    float*           out    = (float*)d_out;

    // workspace carving (floats)
    float* buf0   = (float*)d_ws;                       // ROWS*FP
    float* buf1   = buf0 + (size_t)ROWS * FP;           // ROWS*FP
    float* bufA   = buf1 + (size_t)ROWS * FP;           // ROWS*FP (agg)
    float* Wt     = bufA + (size_t)ROWS * FP;           // KTOT*FP
    float* biasP  = Wt + (size_t)KTOT * FP;             // FP
    float* pooled = biasP + FP;                         // NCH*FP
    float* f1     = pooled + (size_t)NCH * FP;          // 256*128
    float* f2     = f1 + 256 * 128;                     // 256*32
    int*   deg    = (int*)(f2 + 256 * 32);              // N_NODES (pad to 128)
    int*   lists  = deg + 128;                          // N_NODES * N_EDGES

    // stage 0: conv + weight prep + CSR (independent)
    conv1d_grouped<<<ROWS, 256, 0, stream>>>(x, conv_w, conv_b, buf0);
    build_csr<<<1, 128, 0, stream>>>(edges, deg, lists);
    build_wt<<<KTOT, 256, 0, stream>>>(wl, wr, sb, Wt, biasP);

    const dim3 ggrid(ROWS / 512, FP / 32);   // (58, 21)

    // layer 1: buf0 -> buf1
    sage_aggmax<<<ROWS, 256, 0, stream>>>(buf0, deg, lists, bufA);
    sage_gemm<<<ggrid, 256, 0, stream>>>(bufA, buf0, Wt, biasP, buf1);
    // layer 2: buf1 -> buf0
    sage_aggmax<<<ROWS, 256, 0, stream>>>(buf1, deg, lists, bufA);
    sage_gemm<<<ggrid, 256, 0, stream>>>(bufA, buf1, Wt, biasP, buf0);
    // layer 3: buf0 -> buf1
    sage_aggmax<<<ROWS, 256, 0, stream>>>(buf0, deg, lists, bufA);
    sage_gemm<<<ggrid, 256, 0, stream>>>(bufA, buf0, Wt, biasP, buf1);

    // head
    pool_mean<<<NCH, 256, 0, stream>>>(buf1, pooled);
    fc1_k<<<256, 128, 0, stream>>>(pooled, fc1w, fc1b, f1);
    fc2_k<<<256, 32, 0, stream>>>(f1, fc2w, fc2b, f2);
    fc3_softmax<<<256, 4, 0, stream>>>(f2, fc3w, fc3b, out);
}